// Sampler_78297253806173
// MI455X (gfx1250) — compile-verified
//
#include <hip/hip_runtime.h>
#include <hip/hip_bf16.h>

// ---------------------------------------------------------------------------
// Problem constants (static per reference)
// ---------------------------------------------------------------------------
constexpr int Bdim = 4, Cdim = 64, Hdim = 512, Wdim = 512;
constexpr int SIZE  = 16;
constexpr int N_WIN = SIZE * SIZE;        // 256 pixels / window
constexpr int K_SEL = N_WIN / 2;          // k = n * RATIO = 128
constexpr int NH = Hdim / SIZE;           // 32
constexpr int NW = Wdim / SIZE;           // 32
constexpr int HW = Hdim * Wdim;           // 262144
constexpr int NWAVES = N_WIN / 32;        // 8 wave32s per block

typedef __attribute__((ext_vector_type(2))) float v2f;
typedef __attribute__((ext_vector_type(8))) float v8f;

// ---------------------------------------------------------------------------
// wave32 butterfly reductions + 8-wave LDS combine (2 barriers instead of 9)
// ---------------------------------------------------------------------------
__device__ __forceinline__ float wave_sum(float v) {
    #pragma unroll
    for (int off = 16; off > 0; off >>= 1) v += __shfl_xor(v, off, 32);
    return v;
}
__device__ __forceinline__ float wave_max(float v) {
    #pragma unroll
    for (int off = 16; off > 0; off >>= 1) v = fmaxf(v, __shfl_xor(v, off, 32));
    return v;
}

// ---------------------------------------------------------------------------
// Kernel 1: fused attn-conv + entropy top-k mask + masked channel pooling.
// One block per 16x16 window; one thread per pixel.
// Reads a and b exactly once each (512 MB total -> ~22us HBM roofline).
// ---------------------------------------------------------------------------
__global__ __launch_bounds__(N_WIN)
void window_pool_kernel(const float* __restrict__ a,
                        const float* __restrict__ b,
                        const float* __restrict__ attn_w,
                        const float* __restrict__ attn_bias,
                        float* __restrict__ pooled_sum /* [B*C], pre-zeroed */) {
    __shared__ float s_w[Cdim];        // attn weights
    __shared__ float s_win[N_WIN];     // attn values of the window
    __shared__ float s_red[NWAVES];    // per-wave partials
    __shared__ float s_part[Cdim];     // per-channel partial sums

    const int t     = threadIdx.x;               // 0..255 = py*16+px
    const int wid   = t >> 5;                    // wave id (0..7)
    const int lane  = t & 31;
    const int blk   = blockIdx.x;
    const int batch = blk / (NH * NW);
    const int rem   = blk % (NH * NW);
    const int wh    = rem / NW;
    const int ww    = rem % NW;
    const int h     = wh * SIZE + (t >> 4);
    const int w     = ww * SIZE + (t & 15);

    if (t < Cdim) { s_w[t] = attn_w[t]; s_part[t] = 0.0f; }
    __syncthreads();

    // ---- attn = 1x1 conv over channels (coalesced 64B row segments) ----
    const int base = batch * Cdim * HW + h * Wdim + w;
    float acc = attn_bias[0];
    #pragma unroll 4
    for (int c = 0; c < Cdim; ++c) {
        const int cp = (c + 8 < Cdim) ? c + 8 : c;     // branchless clamp
        __builtin_prefetch(&a[base + cp * HW], 0, 0);  // global_prefetch_b8
        acc = fmaf(a[base + c * HW], s_w[c], acc);
    }
    s_win[t] = acc;

    // ---- softmax max (shfl + 8-wave combine) ----
    float wv = wave_max(acc);
    if (lane == 0) s_red[wid] = wv;
    __syncthreads();
    float m = fmaxf(fmaxf(fmaxf(s_red[0], s_red[1]), fmaxf(s_red[2], s_red[3])),
                    fmaxf(fmaxf(s_red[4], s_red[5]), fmaxf(s_red[6], s_red[7])));
    __syncthreads();

    // ---- softmax Z ----
    const float e = expf(acc - m);
    wv = wave_sum(e);
    if (lane == 0) s_red[wid] = wv;
    __syncthreads();
    float Z = ((s_red[0] + s_red[1]) + (s_red[2] + s_red[3])) +
              ((s_red[4] + s_red[5]) + (s_red[6] + s_red[7]));
    __syncthreads();

    // ---- entropy = -sum p*log(p + 1e-8) ----
    const float p = e / Z;
    wv = wave_sum(p * logf(p + 1e-8f));
    if (lane == 0) s_red[wid] = wv;
    __syncthreads();
    const float entropy = -(((s_red[0] + s_red[1]) + (s_red[2] + s_red[3])) +
                            ((s_red[4] + s_red[5]) + (s_red[6] + s_red[7])));

    const float norm_ent = fmaxf(0.1f, entropy / logf((float)N_WIN));
    int k_top = (int)rintf((float)K_SEL * norm_ent);   // round-half-even == jnp.round
    k_top = min(max(k_top, 0), K_SEL);
    const int k_low = K_SEL - k_top;

    // ---- descending rank, stable tie-break (matches double argsort) ----
    int rank = 0;
    const float v = acc;
    for (int j = 0; j < N_WIN; ++j) {                  // broadcast LDS reads
        const float o = s_win[j];
        rank += (o > v) || (o == v && j < t);
    }
    const bool  sel = (rank < k_top) || (rank >= N_WIN - k_low);
    const float mf  = sel ? 1.0f : 0.0f;

    // ---- masked pooling over b: wave32 shuffle reduce -> LDS -> global ----
    for (int c = 0; c < Cdim; ++c) {
        const int cp = (c + 8 < Cdim) ? c + 8 : c;
        __builtin_prefetch(&b[base + cp * HW], 0, 0);
        float val = wave_sum(mf * b[base + c * HW]);
        if (lane == 0) atomicAdd(&s_part[c], val);     // ds_add_f32
    }
    __syncthreads();
    if (t < Cdim) atomicAdd(&pooled_sum[batch * Cdim + t], s_part[t]);
}

// ---------------------------------------------------------------------------
// Kernel 2: tiny MLP with V_WMMA_F32_16X16X4_F32 (single wave32).
//   h   = relu(pooled[4x64 pad 16] @ w1^T[64x16])   -> 16 WMMA, K=64
//   out = h[16x16] @ w2^T[16x64]                    -> 4 N-tiles x 4 WMMA
// A matrix staged zero-padded in LDS so every fragment load is unconditional
// (no EXEC branching between the matrix ops).
// ---------------------------------------------------------------------------
__global__ __launch_bounds__(32)
void mlp_wmma_kernel(const float* __restrict__ pooled_sum,
                     const float* __restrict__ w1,   // [16][64]
                     const float* __restrict__ w2,   // [64][16]
                     float* __restrict__ out) {      // [4][64]
    __shared__ float s_a[16 * 64];   // zero-padded pooled (A of GEMM1)
    __shared__ float s_h[16 * 16];   // relu(h) (A of GEMM2)

    const int L    = threadIdx.x;   // 0..31
    const int half = L >> 4;        // lane half selects K pair (K0/K1 vs K2/K3)
    const int r16  = L & 15;        // M (A) or N (B,C,D) index
    const float scale = 1.0f / (float)HW;   // adaptive avg pool divisor

    // stage A = pooled * scale, rows 4..15 zero (i == row*64+col)
    #pragma unroll
    for (int i = L; i < 16 * 64; i += 32)
        s_a[i] = (i < Bdim * Cdim) ? pooled_sum[i] * scale : 0.0f;
    __syncthreads();

    // ---- GEMM1: D = A_pad(16x64) x B(64x16), B[k][n] = w1[n][k] ----
    v8f c8 = {};
    #pragma unroll
    for (int kc = 0; kc < 16; ++kc) {
        const int k0 = kc * 4 + half * 2;
        v2f av, bv;
        av.x = s_a[r16 * Cdim + k0];
        av.y = s_a[r16 * Cdim + k0 + 1];
        bv.x = w1[r16 * Cdim + k0];
        bv.y = w1[r16 * Cdim + k0 + 1];
        c8 = __builtin_amdgcn_wmma_f32_16x16x4_f32(
                false, av, false, bv, (short)0, c8, false, false);
    }

    // ---- ReLU, spill h to LDS to rebuild A-fragment layout ----
    #pragma unroll
    for (int r = 0; r < 8; ++r) {
        const int M = r + 8 * half;           // C/D row layout
        s_h[M * 16 + r16] = fmaxf(c8[r], 0.0f);
    }
    __syncthreads();

    // ---- GEMM2: out(16x64) = h(16x16) x B, B[k][n] = w2[nt*16+n][k] ----
    #pragma unroll
    for (int nt = 0; nt < 4; ++nt) {
        v8f d8 = {};
        #pragma unroll
        for (int kc = 0; kc < 4; ++kc) {
            const int k0 = kc * 4 + half * 2;
            v2f av, bv;
            av.x = s_h[r16 * 16 + k0];
            av.y = s_h[r16 * 16 + k0 + 1];
            bv.x = w2[(nt * 16 + r16) * 16 + k0];
            bv.y = w2[(nt * 16 + r16) * 16 + k0 + 1];
            d8 = __builtin_amdgcn_wmma_f32_16x16x4_f32(
                    false, av, false, bv, (short)0, d8, false, false);
        }
        if (half == 0) {                       // rows M=0..3 live in lanes 0..15
            #pragma unroll
            for (int r = 0; r < 4; ++r)
                out[r * Cdim + nt * 16 + r16] = d8[r];
        }
    }
}

// ---------------------------------------------------------------------------
extern "C" void kernel_launch(void* const* d_in, const int* in_sizes, int n_in,
                              void* d_out, int out_size, void* d_ws, size_t ws_size,
                              hipStream_t stream) {
    const float* a       = (const float*)d_in[0];
    const float* b       = (const float*)d_in[1];
    const float* attn_w  = (const float*)d_in[2];
    const float* attn_b  = (const float*)d_in[3];
    const float* w1      = (const float*)d_in[4];
    const float* w2      = (const float*)d_in[5];
    // d_in[6] = size (==16), statically known.

    float* pooled = (float*)d_ws;                       // [B*C] accumulator
    hipMemsetAsync(pooled, 0, Bdim * Cdim * sizeof(float), stream);

    window_pool_kernel<<<dim3(Bdim * NH * NW), dim3(N_WIN), 0, stream>>>(
        a, b, attn_w, attn_b, pooled);

    mlp_wmma_kernel<<<dim3(1), dim3(32), 0, stream>>>(
        pooled, w1, w2, (float*)d_out);
}